// GCN_88467736363737
// MI455X (gfx1250) — compile-verified
//
#include <hip/hip_runtime.h>

typedef __attribute__((ext_vector_type(2))) float v2f;
typedef __attribute__((ext_vector_type(8))) float v8f;

#define THREADS 256
#define WAVES_PER_BLOCK (THREADS / 32)

// ---------------------------------------------------------------------------
// Utility: fill a float buffer with a constant (deg=1.0, agg=0.0)
// ---------------------------------------------------------------------------
__global__ void gcn_fill_kernel(float* __restrict__ p, float v, long n) {
  long i = (long)blockIdx.x * blockDim.x + threadIdx.x;
  if (i < n) p[i] = v;
}

// ---------------------------------------------------------------------------
// Degree: deg[c] += 1 for every edge (deg pre-filled with 1.0 for self-loop)
// ---------------------------------------------------------------------------
__global__ void gcn_deg_kernel(const int* __restrict__ col, float* __restrict__ deg, int E) {
  int e = blockIdx.x * blockDim.x + threadIdx.x;
  if (e < E) atomicAdd(&deg[col[e]], 1.0f);
}

// dinv[i] = rsqrt(deg[i])  (deg >= 1 always due to self-loops)
__global__ void gcn_dinv_kernel(const float* __restrict__ deg, float* __restrict__ dinv, int n) {
  int i = blockIdx.x * blockDim.x + threadIdx.x;
  if (i < n) dinv[i] = rsqrtf(deg[i]);
}

// ---------------------------------------------------------------------------
// Zero-pad W3 [64 x 40] -> Wp [64 x 48] so every GEMM has NCOLS % 16 == 0.
// ---------------------------------------------------------------------------
__global__ void gcn_pad_w3_kernel(const float* __restrict__ W, float* __restrict__ Wp) {
  int i = blockIdx.x * blockDim.x + threadIdx.x;
  if (i < 64 * 48) {
    int k = i / 48;
    int c = i - k * 48;
    Wp[i] = (c < 40) ? W[k * 40 + c] : 0.0f;
  }
}

// ---------------------------------------------------------------------------
// GEMM via V_WMMA_F32_16X16X4_F32 (wave32), K and NCOLS compile-time.
// Grid: x = row-tile slabs (8 row tiles / block), y = column tile.
// The K x 16 slab of B for this column tile is staged in LDS once per block
// and shared by all 8 waves. Fully unrolled K-loop: one global_load_b64 (A)
// + two conflict-free ds loads (B) + one WMMA per 4 K-steps, all at
// immediate offsets. No EXEC divergence anywhere (WMMA needs EXEC all-1s).
// ---------------------------------------------------------------------------
template <int K, int NCOLS>
__global__ void gcn_gemm_wmma_kernel(const float* __restrict__ A,
                                     const float* __restrict__ B,
                                     float* __restrict__ C,
                                     int numRowTiles) {
  __shared__ float bs[K * 16];

  int c0 = blockIdx.y << 4;

  // Cooperative stage of B's K x 16 column slab into LDS (coalesced).
#pragma unroll
  for (int i = threadIdx.x; i < K * 16; i += THREADS) {
    int k = i >> 4;
    int c = i & 15;
    bs[i] = B[k * NCOLS + c0 + c];
  }
  __syncthreads();

  int wave = threadIdx.x >> 5;
  int lane = threadIdx.x & 31;
  int rowTile = blockIdx.x * WAVES_PER_BLOCK + wave;
  if (rowTile >= numRowTiles) return;  // wave-uniform exit (after barrier)

  int r0 = rowTile << 4;
  int half = lane >> 4;  // 0: K pair {0,1}; 1: K pair {2,3}
  int l = lane & 15;

  // Lane base for A: row r0+l, starting at K-offset 2*half.
  const float* ap = A + (size_t)(r0 + l) * K + 2 * half;
  // Lane base into LDS B: column l, starting at K-offset 2*half.
  const float* bp = bs + (2 * half) * 16 + l;

  v8f acc = {};
#pragma unroll
  for (int k = 0; k < K; k += 4) {
    v2f a;
    a.x = ap[k];
    a.y = ap[k + 1];
    v2f b;
    b.x = bp[k * 16];
    b.y = bp[k * 16 + 16];
    acc = __builtin_amdgcn_wmma_f32_16x16x4_f32(
        /*neg_a=*/false, a, /*neg_b=*/false, b,
        /*c_mod=*/(short)0, acc, /*reuse_a=*/false, /*reuse_b=*/false);
  }

  // Lane l holds column c0+l; VGPR j holds row r0 + half*8 + j.
  float* cp = C + (size_t)(r0 + half * 8) * NCOLS + c0 + l;
#pragma unroll
  for (int j = 0; j < 8; ++j) cp[j * NCOLS] = acc[j];
}

// ---------------------------------------------------------------------------
// Edge aggregation: agg[col] += dinv[row]*dinv[col] * xw[row]
// One thread per (edge, float4 chunk); consecutive threads cover consecutive
// chunks of one edge so scatter atomics land in 1-2 cachelines per edge.
// All buffers are L2-resident (<<192MB), so these are L2 atomics.
// ld = row stride in floats (64 or 48), d4 = valid float4 chunks per row.
// ---------------------------------------------------------------------------
__global__ void gcn_agg_kernel(const int* __restrict__ row,
                               const int* __restrict__ col,
                               const float* __restrict__ dinv,
                               const float* __restrict__ xw,
                               float* __restrict__ agg,
                               long total, int d4, int ld) {
  long t = (long)blockIdx.x * blockDim.x + threadIdx.x;
  if (t >= total) return;
  int chunk = (int)(t % d4);
  int e = (int)(t / d4);
  int r = row[e];
  int c = col[e];
  float w = dinv[r] * dinv[c];
  const float4* xr = (const float4*)(xw + (size_t)r * ld);
  float4 v = xr[chunk];
  float* op = agg + (size_t)c * ld + (size_t)chunk * 4;
  atomicAdd(op + 0, v.x * w);
  atomicAdd(op + 1, v.y * w);
  atomicAdd(op + 2, v.z * w);
  atomicAdd(op + 3, v.w * w);
}

// ---------------------------------------------------------------------------
// Finalize: out = agg + dinv^2 * xw (self-loop term) + bias, optional ReLU.
// Input buffers have row stride ld; output is dense [N, d].
// ---------------------------------------------------------------------------
__global__ void gcn_finalize_kernel(const float* __restrict__ agg,
                                    const float* __restrict__ xw,
                                    const float* __restrict__ dinv,
                                    const float* __restrict__ bias,
                                    float* __restrict__ out,
                                    long total, int d, int ld, int relu) {
  long t = (long)blockIdx.x * blockDim.x + threadIdx.x;
  if (t >= total) return;
  int node = (int)(t / d);
  int f = (int)(t - (long)node * d);
  size_t in = (size_t)node * ld + f;
  float s = dinv[node];
  float v = agg[in] + s * s * xw[in] + bias[f];
  out[t] = relu ? fmaxf(v, 0.0f) : v;
}

// ---------------------------------------------------------------------------
// Host orchestration
// ---------------------------------------------------------------------------
static inline int cdiv(long a, int b) { return (int)((a + b - 1) / b); }

template <int K, int NCOLS>
static void launch_gemm(const float* A, const float* B, float* C, int M,
                        hipStream_t stream) {
  int numRowTiles = M / 16;  // N = 100000 is divisible by 16
  dim3 grid(cdiv(numRowTiles, WAVES_PER_BLOCK), NCOLS / 16, 1);
  gcn_gemm_wmma_kernel<K, NCOLS><<<grid, THREADS, 0, stream>>>(A, B, C, numRowTiles);
}

extern "C" void kernel_launch(void* const* d_in, const int* in_sizes, int n_in,
                              void* d_out, int out_size, void* d_ws, size_t ws_size,
                              hipStream_t stream) {
  const float* x  = (const float*)d_in[0];
  const int*   ei = (const int*)d_in[1];
  const float* W1 = (const float*)d_in[2];
  const float* b1 = (const float*)d_in[3];
  const float* W2 = (const float*)d_in[4];
  const float* b2 = (const float*)d_in[5];
  const float* W3 = (const float*)d_in[6];
  const float* b3 = (const float*)d_in[7];

  const int D_IN = 128, D_H = 64, D_OUT = 40, D_OUT_PAD = 48;
  int N = in_sizes[0] / D_IN;  // 100000
  int E = in_sizes[1] / 2;     // 1600000
  const int* row = ei;         // edge_index[0]
  const int* col = ei + E;     // edge_index[1]

  // Workspace (floats): deg[N] | dinv[N] | xw[N*64] | agg[N*64] | h[N*64] | wpad[64*48]
  float* ws   = (float*)d_ws;
  float* deg  = ws;
  float* dinv = ws + (size_t)N;
  float* xw   = ws + 2 * (size_t)N;
  float* agg  = xw + (size_t)N * D_H;
  float* h    = agg + (size_t)N * D_H;
  float* wpad = h + (size_t)N * D_H;

  // --- degree / normalization (shared by all 3 layers) ---
  gcn_fill_kernel<<<cdiv(N, THREADS), THREADS, 0, stream>>>(deg, 1.0f, N);
  gcn_deg_kernel<<<cdiv(E, THREADS), THREADS, 0, stream>>>(col, deg, E);
  gcn_dinv_kernel<<<cdiv(N, THREADS), THREADS, 0, stream>>>(deg, dinv, N);
  gcn_pad_w3_kernel<<<cdiv(64 * 48, THREADS), THREADS, 0, stream>>>(W3, wpad);

  // --- layer 1: x[N,128] @ W1 -> xw[N,64]; aggregate; +b, ReLU -> h ---
  {
    const int d = 64, ld = 64, d4 = 16;
    long nd = (long)N * ld, ed = (long)E * d4, od = (long)N * d;
    launch_gemm<128, 64>(x, W1, xw, N, stream);
    gcn_fill_kernel<<<cdiv(nd, THREADS), THREADS, 0, stream>>>(agg, 0.0f, nd);
    gcn_agg_kernel<<<cdiv(ed, THREADS), THREADS, 0, stream>>>(row, col, dinv, xw, agg, ed, d4, ld);
    gcn_finalize_kernel<<<cdiv(od, THREADS), THREADS, 0, stream>>>(agg, xw, dinv, b1, h, od, d, ld, 1);
  }

  // --- layer 2: h @ W2 -> xw; aggregate; +b, ReLU -> h ---
  {
    const int d = 64, ld = 64, d4 = 16;
    long nd = (long)N * ld, ed = (long)E * d4, od = (long)N * d;
    launch_gemm<64, 64>(h, W2, xw, N, stream);
    gcn_fill_kernel<<<cdiv(nd, THREADS), THREADS, 0, stream>>>(agg, 0.0f, nd);
    gcn_agg_kernel<<<cdiv(ed, THREADS), THREADS, 0, stream>>>(row, col, dinv, xw, agg, ed, d4, ld);
    gcn_finalize_kernel<<<cdiv(od, THREADS), THREADS, 0, stream>>>(agg, xw, dinv, b2, h, od, d, ld, 1);
  }

  // --- layer 3: h @ Wpad -> xw[N,48] (cols 40..47 are zero); aggregate;
  //     +b3, no ReLU -> d_out [N,40] dense ---
  {
    const int d = D_OUT, ld = D_OUT_PAD, d4 = 10;
    long nd = (long)N * ld, ed = (long)E * d4, od = (long)N * d;
    launch_gemm<64, 48>(h, wpad, xw, N, stream);
    gcn_fill_kernel<<<cdiv(nd, THREADS), THREADS, 0, stream>>>(agg, 0.0f, nd);
    gcn_agg_kernel<<<cdiv(ed, THREADS), THREADS, 0, stream>>>(row, col, dinv, xw, agg, ed, d4, ld);
    gcn_finalize_kernel<<<cdiv(od, THREADS), THREADS, 0, stream>>>(agg, xw, dinv, b3,
                                                                   (float*)d_out, od, d, ld, 0);
  }
}